// TemporalGNN_11278584119954
// MI455X (gfx1250) — compile-verified
//
#include <hip/hip_runtime.h>
#include <hip/hip_bf16.h>
#include <math.h>

// Problem constants (match reference)
#define NN 200
#define FF 8
#define TT 12
#define OO 32
#define DD (NN * FF * TT)   // 19200
#define EE 1600

typedef __attribute__((ext_vector_type(2))) float v2f;
typedef __attribute__((ext_vector_type(4))) float v4f;
typedef __attribute__((ext_vector_type(8))) float v8f;

// ---------------------------------------------------------------------------
// Kernel 1: xf[row] = dot(x_flat, W_mlp[row, :])  (the 1.47 GB streaming GEMV)
// One 256-thread block per row; non-temporal float4 loads of the W row.
// ---------------------------------------------------------------------------
__global__ void gemv_kernel(const float* __restrict__ x,
                            const float* __restrict__ W,
                            float* __restrict__ xf) {
    __shared__ float red[256];
    const int row = blockIdx.x;
    const v4f* __restrict__ wr = reinterpret_cast<const v4f*>(W + (size_t)row * DD);
    const v4f* __restrict__ xr = reinterpret_cast<const v4f*>(x);
    float acc = 0.0f;
    // DD/4 = 4800 float4 per row
    for (int j = threadIdx.x; j < DD / 4; j += 256) {
        v4f wv = __builtin_nontemporal_load(&wr[j]);  // streamed once: NT hint
        v4f xv = xr[j];                               // hot, keep cached
        acc += wv.x * xv.x + wv.y * xv.y + wv.z * xv.z + wv.w * xv.w;
    }
    red[threadIdx.x] = acc;
    __syncthreads();
    for (int s = 128; s > 0; s >>= 1) {
        if (threadIdx.x < s) red[threadIdx.x] += red[threadIdx.x + s];
        __syncthreads();
    }
    if (threadIdx.x == 0) xf[row] = red[0];
}

// ---------------------------------------------------------------------------
// Kernel 2: GCN normalization degrees (deterministic per-node scan, no atomics)
// deg[n] = 1 (self loop) + sum_{e: dst[e]==n} w[e];  dinv = rsqrt(deg)
// ---------------------------------------------------------------------------
__global__ void dinv_kernel(const int* __restrict__ ei,
                            const float* __restrict__ ew,
                            float* __restrict__ dinv) {
    const int n = blockIdx.x * blockDim.x + threadIdx.x;
    if (n >= NN) return;
    float deg = 1.0f;
    for (int e = 0; e < EE; ++e)
        if (ei[EE + e] == n) deg += ew[e];
    dinv[n] = (deg > 0.0f) ? rsqrtf(deg) : 0.0f;
}

// ---------------------------------------------------------------------------
// Kernel 3: xw[n,t,o] = sum_f x2[n,f,t] * W[f,o]  for both z and h gates
// grid = N, block = T*O = 384
// ---------------------------------------------------------------------------
__global__ void xw_kernel(const float* __restrict__ xf,
                          const float* __restrict__ Wz,
                          const float* __restrict__ Wh,
                          float* __restrict__ xwz,
                          float* __restrict__ xwh) {
    const int n = blockIdx.x;
    const int t = threadIdx.x / OO;   // 0..11
    const int o = threadIdx.x % OO;   // 0..31
    float az = 0.0f, ah = 0.0f;
    for (int f = 0; f < FF; ++f) {
        const float xv = xf[n * (FF * TT) + f * TT + t];
        az += xv * Wz[f * OO + o];
        ah += xv * Wh[f * OO + o];
    }
    const int idx = (n * TT + t) * OO + o;
    xwz[idx] = az;
    xwh[idx] = ah;
}

// ---------------------------------------------------------------------------
// Kernel 4: message gather (deterministic order): C[n] = dinv[n]^2*xw[n]
//           + sum_{e: dst==n} dinv[src]*w*dinv[n] * xw[src]
// Edge list staged in LDS. grid = N, block = 384
// ---------------------------------------------------------------------------
__global__ void gather_kernel(const int* __restrict__ ei,
                              const float* __restrict__ ew,
                              const float* __restrict__ dinv,
                              const float* __restrict__ xwz,
                              const float* __restrict__ xwh,
                              float* __restrict__ Cz,
                              float* __restrict__ Ch) {
    __shared__ int   s_src[EE];
    __shared__ int   s_dst[EE];
    __shared__ float s_w[EE];
    const int n = blockIdx.x;
    for (int e = threadIdx.x; e < EE; e += 384) {
        s_src[e] = ei[e];
        s_dst[e] = ei[EE + e];
        s_w[e]   = ew[e];
    }
    __syncthreads();
    const int t = threadIdx.x / OO;
    const int o = threadIdx.x % OO;
    const float dn = dinv[n];
    const int   idx = (n * TT + t) * OO + o;
    float az = dn * dn * xwz[idx];   // self loop, weight 1
    float ah = dn * dn * xwh[idx];
    for (int e = 0; e < EE; ++e) {
        if (s_dst[e] == n) {
            const int   s   = s_src[e];
            const float nrm = dinv[s] * s_w[e] * dn;
            const int   si  = (s * TT + t) * OO + o;
            az += nrm * xwz[si];
            ah += nrm * xwh[si];
        }
    }
    Cz[idx] = az;
    Ch[idx] = ah;
}

// ---------------------------------------------------------------------------
// Kernel 5: gate GEMMs via V_WMMA_F32_16X16X4_F32 (exact fp32).
//   Zpre[m,p] = sum_o Cz[m,o]*Lz[p,o] + blz[p]   (m = n*T+t, M = 2400)
//   Hpre[m,p] = sum_o Ch[m,o]*Lh[p,o] + blh[p]
//   Hp = (1 - sigmoid(Zpre)) * tanh(Hpre)
// One wave per 16-row M tile; 8 chained K=4 WMMAs per 16x16 tile,
// 2 p-tiles x 2 gates = 32 WMMAs per wave. EXEC kept all-ones (wave-uniform
// guard only), per ISA WMMA restriction.
// A layout (16x4 f32): lanes 0-15 -> K 0/1, lanes 16-31 -> K 2/3, M = lane&15.
// B layout (4x16 f32): symmetric, N = lane&15.
// C/D layout: VGPR v -> M = v (+8 for upper lanes), N = lane&15.
// ---------------------------------------------------------------------------
__global__ void gate_wmma_kernel(const float* __restrict__ Cz,
                                 const float* __restrict__ Ch,
                                 const float* __restrict__ Lz,
                                 const float* __restrict__ blz,
                                 const float* __restrict__ Lh,
                                 const float* __restrict__ blh,
                                 float* __restrict__ Hp) {
    const int wave = (blockIdx.x * blockDim.x + threadIdx.x) >> 5;
    const int lane = threadIdx.x & 31;
    if (wave >= (NN * TT) / 16) return;   // wave-uniform; EXEC stays all-1s
    const int mbase = wave * 16;
    const int r     = lane & 15;          // row within tile (A) / col (B,D)
    const int khalf = (lane >> 4) << 1;   // 0 for lanes 0-15, 2 for 16-31

    for (int pt = 0; pt < 2; ++pt) {
        const int p = pt * 16 + r;        // output column this lane owns
        v8f cz, ch;
        const float bz_v = blz[p];
        const float bh_v = blh[p];
#pragma unroll
        for (int v = 0; v < 8; ++v) { cz[v] = bz_v; ch[v] = bh_v; }

#pragma unroll
        for (int kb = 0; kb < 8; ++kb) {
            const int k0 = kb * 4 + khalf;
            v2f a, b;
            // z gate
            a.x = Cz[(mbase + r) * OO + k0];
            a.y = Cz[(mbase + r) * OO + k0 + 1];
            b.x = Lz[p * (2 * OO) + k0];       // Lz is (O, 2O); use [:, :O]
            b.y = Lz[p * (2 * OO) + k0 + 1];
            cz = __builtin_amdgcn_wmma_f32_16x16x4_f32(
                     false, a, false, b, (short)0, cz, false, false);
            // h gate
            a.x = Ch[(mbase + r) * OO + k0];
            a.y = Ch[(mbase + r) * OO + k0 + 1];
            b.x = Lh[p * (2 * OO) + k0];
            b.y = Lh[p * (2 * OO) + k0 + 1];
            ch = __builtin_amdgcn_wmma_f32_16x16x4_f32(
                     false, a, false, b, (short)0, ch, false, false);
        }

#pragma unroll
        for (int v = 0; v < 8; ++v) {
            const float z  = 1.0f / (1.0f + expf(-cz[v]));
            const float ht = tanhf(ch[v]);
            const float hp = (1.0f - z) * ht;
            const int m = mbase + ((lane < 16) ? v : (v + 8));
            Hp[m * OO + p] = hp;
        }
    }
}

// ---------------------------------------------------------------------------
// Kernel 6: attention over periods + output head.
// H[n,p] = sum_t Hp[n*T+t, p] * softmax(att)[t]; h = relu(H);
// out[n,tt] = sum_p h[p] * W_out[tt,p] + b_out[tt]
// grid = N, block = 32
// ---------------------------------------------------------------------------
__global__ void out_kernel(const float* __restrict__ Hp,
                           const float* __restrict__ att,
                           const float* __restrict__ W_out,
                           const float* __restrict__ b_out,
                           float* __restrict__ out) {
    __shared__ float h[OO];
    const int n = blockIdx.x;
    const int p = threadIdx.x;   // 0..31
    // softmax over att (redundant per thread, trivial, deterministic)
    float a[TT], m = -1e30f, s = 0.0f;
    for (int t = 0; t < TT; ++t) { a[t] = att[t]; m = fmaxf(m, a[t]); }
    for (int t = 0; t < TT; ++t) { a[t] = expf(a[t] - m); s += a[t]; }
    const float inv = 1.0f / s;
    float acc = 0.0f;
    for (int t = 0; t < TT; ++t) acc += Hp[(n * TT + t) * OO + p] * (a[t] * inv);
    h[p] = fmaxf(acc, 0.0f);
    __syncthreads();
    if (p < TT) {
        float o = b_out[p];
        for (int q = 0; q < OO; ++q) o += h[q] * W_out[p * OO + q];
        out[n * TT + p] = o;
    }
}

// ---------------------------------------------------------------------------
// Host-side launcher
// Inputs (setup_inputs order):
//  0 x(N,F,T) f32   1 edge_index(2,E) i32   2 edge_weight(E) f32
//  3 W_mlp(D,D) f32 4 Wz  5 bz  6 Wr  7 br  8 Wh  9 bh
// 10 Lz 11 blz 12 Lr 13 blr 14 Lh 15 blh 16 att 17 W_out 18 b_out
// ---------------------------------------------------------------------------
extern "C" void kernel_launch(void* const* d_in, const int* in_sizes, int n_in,
                              void* d_out, int out_size, void* d_ws, size_t ws_size,
                              hipStream_t stream) {
    const float* x     = (const float*)d_in[0];
    const int*   ei    = (const int*)  d_in[1];
    const float* ew    = (const float*)d_in[2];
    const float* Wmlp  = (const float*)d_in[3];
    const float* Wz    = (const float*)d_in[4];
    const float* Wh    = (const float*)d_in[8];
    const float* Lz    = (const float*)d_in[10];
    const float* blz   = (const float*)d_in[11];
    const float* Lh    = (const float*)d_in[14];
    const float* blh   = (const float*)d_in[15];
    const float* att   = (const float*)d_in[16];
    const float* Wout  = (const float*)d_in[17];
    const float* bout  = (const float*)d_in[18];
    float* out = (float*)d_out;

    // workspace layout (floats)
    float* ws   = (float*)d_ws;
    float* xf   = ws;                       // 19200
    float* dinv = ws + 19200;               // 200 (pad to 256)
    float* xwz  = ws + 19456;               // 76800
    float* xwh  = ws + 19456 + 76800;       // 76800
    float* Cz   = ws + 19456 + 2 * 76800;   // 76800
    float* Ch   = ws + 19456 + 3 * 76800;   // 76800
    float* Hp   = ws + 19456 + 4 * 76800;   // 76800  (total ~1.62 MB)

    gemv_kernel<<<DD, 256, 0, stream>>>(x, Wmlp, xf);
    dinv_kernel<<<1, 256, 0, stream>>>(ei, ew, dinv);
    xw_kernel<<<NN, TT * OO, 0, stream>>>(xf, Wz, Wh, xwz, xwh);
    gather_kernel<<<NN, TT * OO, 0, stream>>>(ei, ew, dinv, xwz, xwh, Cz, Ch);
    // 150 M-tiles; 128-thread blocks = 4 waves each -> 38 blocks
    gate_wmma_kernel<<<38, 128, 0, stream>>>(Cz, Ch, Lz, blz, Lh, blh, Hp);
    out_kernel<<<NN, 32, 0, stream>>>(Hp, att, Wout, bout, out);
}